// HybridGNN_4002909520224
// MI455X (gfx1250) — compile-verified
//
#include <hip/hip_runtime.h>
#include <hip/hip_bf16.h>

#define NN      100000
#define EE      300000
#define F_IN    128
#define HID     256
#define HEADS   4
#define HC      64
#define GG      4000
#define BN_EPS  1e-5f

typedef __attribute__((ext_vector_type(2))) float v2f;
typedef __attribute__((ext_vector_type(8))) float v8f;
typedef __attribute__((ext_vector_type(4))) int   v4i;

#if defined(__gfx1250__) && __has_builtin(__builtin_amdgcn_global_load_async_to_lds_b128)
#define HAS_ASYNC 1
#else
#define HAS_ASYNC 0
#endif

#if __has_builtin(__builtin_amdgcn_s_wait_asynccnt)
#define WAIT_ASYNC(n) __builtin_amdgcn_s_wait_asynccnt(n)
#else
#define WAIT_ASYNC(n) asm volatile("s_wait_asynccnt %0" :: "i"(n) : "memory")
#endif

// ---------------------------------------------------------------------------
// WMMA fp32 GEMM:  C[M,Nout] = act(A[M,K] @ W[K,Nout] (+ bias))
// block = 256 threads = 8 waves; block tile 32 rows x 128 cols.
// wave tile 16x32 (two 16x16 accumulators sharing the A fragment).
// B staged in LDS in K-chunks of 16, double buffered, async-loaded.
// grid = (M/32, Nout/128). M%32==0, Nout%128==0, K%16==0 by construction.
// ---------------------------------------------------------------------------
#define KC 16
#define SP 136   // padded LDS row stride (floats): conflict-free fragment reads

__global__ __launch_bounds__(256)
void gemm_wmma(const float* __restrict__ A, const float* __restrict__ W,
               const float* __restrict__ bias, float* __restrict__ C,
               int M, int K, int Nout, int do_relu, int do_bias)
{
    __shared__ float sB[2][KC * SP];

    const int tid  = threadIdx.x;
    const int lane = tid & 31;
    const int wave = tid >> 5;
    const int half = lane >> 4;          // 0: K+{0,1}, 1: K+{2,3}
    const int l16  = lane & 15;
    const int row0 = blockIdx.x * 32 + (wave & 1) * 16;
    const int colB = blockIdx.y * 128;

    // per-thread B staging: two float4 per chunk (2048 floats / 256 threads)
    const int r0s = tid >> 5,        c0s = (tid & 31) * 4;          // ids 0..255
    const int r1s = (tid >> 5) + 8,  c1s = c0s;                     // ids 256..511

    auto issue = [&](int k0, int buf) {
#if HAS_ASYNC
        __builtin_amdgcn_global_load_async_to_lds_b128(
            (v4i*)(W + (size_t)(k0 + r0s) * Nout + colB + c0s),
            (v4i*)&sB[buf][r0s * SP + c0s], 0, 0);
        __builtin_amdgcn_global_load_async_to_lds_b128(
            (v4i*)(W + (size_t)(k0 + r1s) * Nout + colB + c1s),
            (v4i*)&sB[buf][r1s * SP + c1s], 0, 0);
#else
        *(float4*)&sB[buf][r0s * SP + c0s] =
            *(const float4*)(W + (size_t)(k0 + r0s) * Nout + colB + c0s);
        *(float4*)&sB[buf][r1s * SP + c1s] =
            *(const float4*)(W + (size_t)(k0 + r1s) * Nout + colB + c1s);
#endif
    };

    v8f c0 = {}, c1 = {};
    const float* __restrict__ Arow = A + (size_t)(row0 + l16) * K;
    const int koff = 2 * half;
    const int cA = (wave >> 1) * 32 + l16;   // tile-0 col within block tile
    const int cB = cA + 16;                  // tile-1 col

    issue(0, 0);
    const int nch = K / KC;
    for (int ch = 0; ch < nch; ++ch) {
        const int k0 = ch * KC;
        const bool more = (ch + 1) < nch;
        if (more) issue(k0 + KC, (ch + 1) & 1);
#if HAS_ASYNC
        if (more) { WAIT_ASYNC(2); } else { WAIT_ASYNC(0); }
#endif
        __syncthreads();
        const float* sb = sB[ch & 1];
        if (more) __builtin_prefetch(Arow + k0 + KC, 0, 0);
#pragma unroll
        for (int kk = 0; kk < KC; kk += 4) {
            const int rb = kk + koff;
            v2f a = *(const v2f*)(Arow + k0 + rb);
            v2f b0, b1;
            b0.x = sb[rb * SP + cA];
            b0.y = sb[(rb + 1) * SP + cA];
            b1.x = sb[rb * SP + cB];
            b1.y = sb[(rb + 1) * SP + cB];
            // 8 args: (neg_a, A, neg_b, B, c_mod, C, reuse_a, reuse_b)
            c0 = __builtin_amdgcn_wmma_f32_16x16x4_f32(false, a, false, b0,
                                                       (short)0, c0, false, false);
            c1 = __builtin_amdgcn_wmma_f32_16x16x4_f32(false, a, false, b1,
                                                       (short)0, c1, false, false);
        }
        __syncthreads();
    }

    const int gc0 = colB + cA;
    const int gc1 = colB + cB;
    const float bv0 = do_bias ? bias[gc0] : 0.0f;
    const float bv1 = do_bias ? bias[gc1] : 0.0f;
#pragma unroll
    for (int i = 0; i < 8; ++i) {
        const int m = row0 + i + 8 * half;
        float v0 = c0[i] + bv0;
        float v1 = c1[i] + bv1;
        if (do_relu) { v0 = fmaxf(v0, 0.0f); v1 = fmaxf(v1, 0.0f); }
        C[(size_t)m * Nout + gc0] = v0;
        C[(size_t)m * Nout + gc1] = v1;
    }
}

// ---------------------------------------------------------------------------
// elementwise helpers
// ---------------------------------------------------------------------------
__global__ void k_fill(float* p, float v, size_t n) {
    size_t t = (size_t)blockIdx.x * blockDim.x + threadIdx.x;
    if (t < n) p[t] = v;
}
__global__ void k_copy(float* d, const float* s, size_t n) {
    size_t t = (size_t)blockIdx.x * blockDim.x + threadIdx.x;
    if (t < n) d[t] = s[t];
}
__global__ void k_bias_init(float* out, const float* bias, size_t n) {
    size_t t = (size_t)blockIdx.x * blockDim.x + threadIdx.x;
    if (t < n) out[t] = bias[t & (HID - 1)];
}

// ---------------------------------------------------------------------------
// degrees / GCN
// ---------------------------------------------------------------------------
__global__ void k_deg_acc(float* deg, const int* dst) {
    int t = blockIdx.x * blockDim.x + threadIdx.x;
    if (t < EE) atomicAdd(&deg[dst[t]], 1.0f);
}
__global__ void k_dinv(const float* deg, float* dinv) {
    int t = blockIdx.x * blockDim.x + threadIdx.x;
    if (t < NN) dinv[t] = rsqrtf(deg[t]);   // deg >= 1 (self loop)
}
// out = bias + dinv[n]^2 * h   (self-loop contribution of GCNConv)
__global__ void k_gcn_init(float* out, const float* h, const float* dinv,
                           const float* bias) {
    size_t t = (size_t)blockIdx.x * blockDim.x + threadIdx.x;
    if (t >= (size_t)NN * HID) return;
    int n = (int)(t >> 8), c = (int)(t & (HID - 1));
    float di = dinv[n];
    out[t] = bias[c] + h[t] * di * di;
}
__global__ void k_gcn_scatter(float* out, const float* h, const float* dinv,
                              const int* src, const int* dst) {
    size_t t = (size_t)blockIdx.x * blockDim.x + threadIdx.x;
    if (t >= (size_t)EE * (HID / 4)) return;
    int chunk = (int)(t & 63);
    int e     = (int)(t >> 6);
    int s = src[e], d = dst[e];
    float w = dinv[s] * dinv[d];
    const float4 hv = *(const float4*)(h + (size_t)s * HID + chunk * 4);
    float* op = out + (size_t)d * HID + chunk * 4;
    atomicAdd(op + 0, hv.x * w);
    atomicAdd(op + 1, hv.y * w);
    atomicAdd(op + 2, hv.z * w);
    atomicAdd(op + 3, hv.w * w);
}

// ---------------------------------------------------------------------------
// GAT
// ---------------------------------------------------------------------------
__device__ __forceinline__ unsigned fkey(float f) {
    unsigned u = __float_as_uint(f);
    return (u & 0x80000000u) ? ~u : (u | 0x80000000u);
}
__device__ __forceinline__ float unfkey(unsigned u) {
    return (u & 0x80000000u) ? __uint_as_float(u & 0x7fffffffu)
                             : __uint_as_float(~u);
}
__device__ __forceinline__ float lrelu(float v) {
    return v > 0.0f ? v : 0.2f * v;
}
__device__ __forceinline__ void att_edge(int e, const int* src, const int* dst,
                                         int& s, int& d) {
    if (e < EE) { s = src[e]; d = dst[e]; }
    else        { s = e - EE; d = e - EE; }   // self loop
}

__global__ void k_att_logits(const float* H, const float* aS, const float* aD,
                             float* alS, float* alD) {
    int t = blockIdx.x * blockDim.x + threadIdx.x;
    if (t >= NN * HEADS) return;
    int n = t >> 2, h = t & 3;
    const float* hp = H + (size_t)n * HID + h * HC;
    const float* as = aS + h * HC;
    const float* ad = aD + h * HC;
    float s1 = 0.f, s2 = 0.f;
    for (int c = 0; c < HC; ++c) { float v = hp[c]; s1 += v * as[c]; s2 += v * ad[c]; }
    alS[t] = s1; alD[t] = s2;
}
__global__ void k_att_max(const float* alS, const float* alD, unsigned* keys,
                          const int* src, const int* dst) {
    int t = blockIdx.x * blockDim.x + threadIdx.x;
    if (t >= (EE + NN) * HEADS) return;
    int h = t & 3, e = t >> 2, s, d;
    att_edge(e, src, dst, s, d);
    float ev = lrelu(alS[s * HEADS + h] + alD[d * HEADS + h]);
    atomicMax(&keys[d * HEADS + h], fkey(ev));
}
__global__ void k_att_denom(const float* alS, const float* alD,
                            const unsigned* keys, float* denom,
                            const int* src, const int* dst) {
    int t = blockIdx.x * blockDim.x + threadIdx.x;
    if (t >= (EE + NN) * HEADS) return;
    int h = t & 3, e = t >> 2, s, d;
    att_edge(e, src, dst, s, d);
    float ev = lrelu(alS[s * HEADS + h] + alD[d * HEADS + h]);
    float ee = __expf(ev - unfkey(keys[d * HEADS + h]));
    atomicAdd(&denom[d * HEADS + h], ee);
}
__global__ void k_att_scatter(const float* H, const float* alS, const float* alD,
                              const unsigned* keys, const float* denom,
                              float* out, const int* src, const int* dst) {
    size_t t = (size_t)blockIdx.x * blockDim.x + threadIdx.x;
    if (t >= (size_t)(EE + NN) * HEADS * (HC / 4)) return;
    int chunk = (int)(t & 15);
    int h     = (int)((t >> 4) & 3);
    int e     = (int)(t >> 6);
    int s, d;
    att_edge(e, src, dst, s, d);
    float ev = lrelu(alS[s * HEADS + h] + alD[d * HEADS + h]);
    float ee = __expf(ev - unfkey(keys[d * HEADS + h]));
    float alpha = ee / (denom[d * HEADS + h] + 1e-16f);
    const float4 hv = *(const float4*)(H + (size_t)s * HID + h * HC + chunk * 4);
    float* op = out + (size_t)d * HID + h * HC + chunk * 4;
    atomicAdd(op + 0, hv.x * alpha);
    atomicAdd(op + 1, hv.y * alpha);
    atomicAdd(op + 2, hv.z * alpha);
    atomicAdd(op + 3, hv.w * alpha);
}

// ---------------------------------------------------------------------------
// GIN scatter:  out[dst] += h0[src]   (out pre-initialized to h0)
// ---------------------------------------------------------------------------
__global__ void k_gin_scatter(float* out, const float* h0,
                              const int* src, const int* dst) {
    size_t t = (size_t)blockIdx.x * blockDim.x + threadIdx.x;
    if (t >= (size_t)EE * (HID / 4)) return;
    int chunk = (int)(t & 63);
    int e     = (int)(t >> 6);
    int s = src[e], d = dst[e];
    const float4 hv = *(const float4*)(h0 + (size_t)s * HID + chunk * 4);
    float* op = out + (size_t)d * HID + chunk * 4;
    atomicAdd(op + 0, hv.x);
    atomicAdd(op + 1, hv.y);
    atomicAdd(op + 2, hv.z);
    atomicAdd(op + 3, hv.w);
}

// ---------------------------------------------------------------------------
// BatchNorm (C == HID == 256): per-channel sum / sumsq, then apply
// ---------------------------------------------------------------------------
__global__ __launch_bounds__(256)
void k_bn_stats(const float* X, int M, float* sum, float* sumsq) {
    int c  = threadIdx.x;                  // 256 channels
    int r0 = blockIdx.x * 256;
    int r1 = r0 + 256; if (r1 > M) r1 = M;
    float s = 0.f, s2 = 0.f;
    for (int r = r0; r < r1; ++r) {
        float v = X[(size_t)r * HID + c];
        s += v; s2 += v * v;
    }
    atomicAdd(&sum[c], s);
    atomicAdd(&sumsq[c], s2);
}
__global__ void k_bn_apply(const float* X, float* Y, const float* sum,
                           const float* sumsq, const float* g, const float* b,
                           int M, int do_relu, size_t total) {
    size_t t = (size_t)blockIdx.x * blockDim.x + threadIdx.x;
    if (t >= total) return;
    int c = (int)(t & (HID - 1));
    float m = sum[c] / (float)M;
    float v = sumsq[c] / (float)M - m * m;
    float y = (X[t] - m) * rsqrtf(v + BN_EPS) * g[c] + b[c];
    if (do_relu) y = fmaxf(y, 0.f);
    Y[t] = y;
}

// ---------------------------------------------------------------------------
// mean pooling into pooled[G, 3*HID]
// ---------------------------------------------------------------------------
__global__ void k_pool_cnt(const int* batch, float* cnt) {
    int t = blockIdx.x * blockDim.x + threadIdx.x;
    if (t < NN) atomicAdd(&cnt[batch[t]], 1.0f);
}
__global__ void k_pool_acc(const float* X, const int* batch, float* pooled,
                           int off) {
    size_t t = (size_t)blockIdx.x * blockDim.x + threadIdx.x;
    if (t >= (size_t)NN * (HID / 4)) return;
    int chunk = (int)(t & 63);
    int n     = (int)(t >> 6);
    int g = batch[n];
    const float4 v = *(const float4*)(X + (size_t)n * HID + chunk * 4);
    float* op = pooled + (size_t)g * (3 * HID) + off + chunk * 4;
    atomicAdd(op + 0, v.x);
    atomicAdd(op + 1, v.y);
    atomicAdd(op + 2, v.z);
    atomicAdd(op + 3, v.w);
}
__global__ void k_pool_div(float* pooled, const float* cnt) {
    int t = blockIdx.x * blockDim.x + threadIdx.x;
    if (t >= GG * 3 * HID) return;
    pooled[t] /= fmaxf(cnt[t / (3 * HID)], 1.0f);
}

// ---------------------------------------------------------------------------
// classifier head:  out[g] = g2[g,:] . clsW + cls_b
// ---------------------------------------------------------------------------
__global__ void k_cls(const float* g2, const float* clsW, const float* clsb,
                      float* out) {
    int t = blockIdx.x * blockDim.x + threadIdx.x;
    if (t >= GG) return;
    const float* row = g2 + (size_t)t * (HID / 2);
    float s = 0.f;
    for (int c = 0; c < HID / 2; ++c) s += row[c] * clsW[c];
    out[t] = s + clsb[0];
}

// ---------------------------------------------------------------------------
extern "C" void kernel_launch(void* const* d_in, const int* in_sizes, int n_in,
                              void* d_out, int out_size, void* d_ws, size_t ws_size,
                              hipStream_t stream) {
    const float* x   = (const float*)d_in[0];
    const int*   ei  = (const int*)d_in[1];
    const int*   bat = (const int*)d_in[2];
    const int* src = ei;
    const int* dst = ei + EE;

    // params in _init_params dict order, starting at d_in[3]
    const float* proj_W = (const float*)d_in[3];
    const float* proj_b = (const float*)d_in[4];
    const float* gcn1_W = (const float*)d_in[5];
    const float* gcn1_b = (const float*)d_in[6];
    const float* gcn2_W = (const float*)d_in[7];
    const float* gcn2_b = (const float*)d_in[8];
    const float* gat1_W    = (const float*)d_in[9];
    const float* gat1_asrc = (const float*)d_in[10];
    const float* gat1_adst = (const float*)d_in[11];
    const float* gat1_b    = (const float*)d_in[12];
    const float* gat2_W    = (const float*)d_in[13];
    const float* gat2_asrc = (const float*)d_in[14];
    const float* gat2_adst = (const float*)d_in[15];
    const float* gat2_b    = (const float*)d_in[16];
    const float* gin_W1 = (const float*)d_in[17];
    const float* gin_b1 = (const float*)d_in[18];
    const float* gin_W2 = (const float*)d_in[19];
    const float* gin_b2 = (const float*)d_in[20];
    const float* bn1_g = (const float*)d_in[21];
    const float* bn1_b = (const float*)d_in[22];
    const float* bn2_g = (const float*)d_in[23];
    const float* bn2_b = (const float*)d_in[24];
    const float* bn3_g = (const float*)d_in[25];
    const float* bn3_b = (const float*)d_in[26];
    const float* bng_g = (const float*)d_in[27];
    const float* bng_b = (const float*)d_in[28];
    const float* fc1_W = (const float*)d_in[29];
    const float* fc1_b = (const float*)d_in[30];
    const float* fc2_W = (const float*)d_in[31];
    const float* fc2_b = (const float*)d_in[32];
    const float* cls_W = (const float*)d_in[33];
    const float* cls_b = (const float*)d_in[34];
    float* out = (float*)d_out;

    // ---- workspace layout (floats) ----
    float* F = (float*)d_ws;
    size_t o = 0;
    float* h0  = F + o; o += (size_t)NN * HID;
    float* B1  = F + o; o += (size_t)NN * HID;
    float* B2  = F + o; o += (size_t)NN * HID;
    float* B3  = F + o; o += (size_t)NN * 2 * HID;   // GIN hidden [N,512]
    float* deg = F + o; o += NN;
    float* din = F + o; o += NN;
    float* alS = F + o; o += (size_t)NN * HEADS;
    float* alD = F + o; o += (size_t)NN * HEADS;
    unsigned* keys = (unsigned*)(F + o); o += (size_t)NN * HEADS;
    float* den = F + o; o += (size_t)NN * HEADS;
    float* pooled = F + o; o += (size_t)GG * 3 * HID;
    float* cnt = F + o; o += GG;
    float* bnsum = F + o; o += HID;
    float* bnsq  = F + o; o += HID;
    float* g1 = F + o; o += (size_t)GG * HID;
    float* g2 = F + o; o += (size_t)GG * (HID / 2);
    (void)ws_size; (void)n_in; (void)in_sizes; (void)out_size;

    auto gemm = [&](const float* A, const float* W, const float* bias, float* C,
                    int M, int K, int Nout, int relu, int hasb) {
        dim3 grid(M / 32, Nout / 128);
        gemm_wmma<<<grid, 256, 0, stream>>>(A, W, bias, C, M, K, Nout, relu, hasb);
    };
    auto g1d = [](size_t n) { return dim3((unsigned)((n + 255) / 256)); };

    const size_t NH = (size_t)NN * HID;

    // h0 = relu(x @ proj_W + proj_b)
    gemm(x, proj_W, proj_b, h0, NN, F_IN, HID, 1, 1);

    // degrees (self-loop augmented) and dinv
    k_fill<<<g1d(NN), 256, 0, stream>>>(deg, 1.0f, NN);
    k_deg_acc<<<g1d(EE), 256, 0, stream>>>(deg, dst);
    k_dinv<<<g1d(NN), 256, 0, stream>>>(deg, din);

    // pooling accumulators
    k_fill<<<g1d((size_t)GG * 3 * HID), 256, 0, stream>>>(pooled, 0.f, (size_t)GG * 3 * HID);
    k_fill<<<g1d(GG), 256, 0, stream>>>(cnt, 0.f, GG);
    k_pool_cnt<<<g1d(NN), 256, 0, stream>>>(bat, cnt);

    // ================= GCN branch =================
    gemm(h0, gcn1_W, nullptr, B1, NN, HID, HID, 0, 0);
    k_gcn_init<<<g1d(NH), 256, 0, stream>>>(B2, B1, din, gcn1_b);
    k_gcn_scatter<<<g1d((size_t)EE * 64), 256, 0, stream>>>(B2, B1, din, src, dst);
    k_fill<<<g1d(2 * HID), 256, 0, stream>>>(bnsum, 0.f, 2 * HID);
    k_bn_stats<<<(NN + 255) / 256, 256, 0, stream>>>(B2, NN, bnsum, bnsq);
    k_bn_apply<<<g1d(NH), 256, 0, stream>>>(B2, B2, bnsum, bnsq, bn1_g, bn1_b, NN, 1, NH);
    gemm(B2, gcn2_W, nullptr, B1, NN, HID, HID, 0, 0);
    k_gcn_init<<<g1d(NH), 256, 0, stream>>>(B2, B1, din, gcn2_b);
    k_gcn_scatter<<<g1d((size_t)EE * 64), 256, 0, stream>>>(B2, B1, din, src, dst);
    k_pool_acc<<<g1d((size_t)NN * 64), 256, 0, stream>>>(B2, bat, pooled, 0);

    // ================= GAT branch =================
    for (int layer = 0; layer < 2; ++layer) {
        const float* in  = (layer == 0) ? h0 : B2;
        const float* W   = (layer == 0) ? gat1_W : gat2_W;
        const float* as  = (layer == 0) ? gat1_asrc : gat2_asrc;
        const float* ad  = (layer == 0) ? gat1_adst : gat2_adst;
        const float* bb  = (layer == 0) ? gat1_b : gat2_b;
        gemm(in, W, nullptr, B1, NN, HID, HID, 0, 0);
        k_att_logits<<<g1d((size_t)NN * HEADS), 256, 0, stream>>>(B1, as, ad, alS, alD);
        k_bias_init<<<g1d(NH), 256, 0, stream>>>(B2, bb, NH);
        k_fill<<<g1d((size_t)NN * HEADS), 256, 0, stream>>>((float*)keys, 0.f, (size_t)NN * HEADS);
        k_fill<<<g1d((size_t)NN * HEADS), 256, 0, stream>>>(den, 0.f, (size_t)NN * HEADS);
        k_att_max<<<g1d((size_t)(EE + NN) * HEADS), 256, 0, stream>>>(alS, alD, keys, src, dst);
        k_att_denom<<<g1d((size_t)(EE + NN) * HEADS), 256, 0, stream>>>(alS, alD, keys, den, src, dst);
        k_att_scatter<<<g1d((size_t)(EE + NN) * HEADS * 16), 256, 0, stream>>>(
            B1, alS, alD, keys, den, B2, src, dst);
        if (layer == 0) {
            k_fill<<<g1d(2 * HID), 256, 0, stream>>>(bnsum, 0.f, 2 * HID);
            k_bn_stats<<<(NN + 255) / 256, 256, 0, stream>>>(B2, NN, bnsum, bnsq);
            k_bn_apply<<<g1d(NH), 256, 0, stream>>>(B2, B2, bnsum, bnsq, bng_g, bng_b, NN, 1, NH);
        }
    }
    k_pool_acc<<<g1d((size_t)NN * 64), 256, 0, stream>>>(B2, bat, pooled, HID);

    // ================= GIN branch =================
    k_copy<<<g1d(NH), 256, 0, stream>>>(B1, h0, NH);             // h = x + agg (identity term)
    k_gin_scatter<<<g1d((size_t)EE * 64), 256, 0, stream>>>(B1, h0, src, dst);
    gemm(B1, gin_W1, gin_b1, B3, NN, HID, 2 * HID, 1, 1);        // relu(h@W1+b1)
    gemm(B3, gin_W2, gin_b2, B1, NN, 2 * HID, HID, 0, 1);        // @W2+b2
    k_fill<<<g1d(2 * HID), 256, 0, stream>>>(bnsum, 0.f, 2 * HID);
    k_bn_stats<<<(NN + 255) / 256, 256, 0, stream>>>(B1, NN, bnsum, bnsq);
    k_bn_apply<<<g1d(NH), 256, 0, stream>>>(B1, B1, bnsum, bnsq, bn2_g, bn2_b, NN, 1, NH);
    k_pool_acc<<<g1d((size_t)NN * 64), 256, 0, stream>>>(B1, bat, pooled, 2 * HID);

    // ================= head =================
    k_pool_div<<<g1d((size_t)GG * 3 * HID), 256, 0, stream>>>(pooled, cnt);
    gemm(pooled, fc1_W, fc1_b, g1, GG, 3 * HID, HID, 0, 1);
    k_fill<<<g1d(2 * HID), 256, 0, stream>>>(bnsum, 0.f, 2 * HID);
    k_bn_stats<<<(GG + 255) / 256, 256, 0, stream>>>(g1, GG, bnsum, bnsq);
    k_bn_apply<<<g1d((size_t)GG * HID), 256, 0, stream>>>(g1, g1, bnsum, bnsq, bn3_g, bn3_b, GG, 1, (size_t)GG * HID);
    gemm(g1, fc2_W, fc2_b, g2, GG, HID, HID / 2, 1, 1);
    k_cls<<<g1d(GG), 256, 0, stream>>>(g2, cls_W, cls_b, out);
}